// Model_82042465289182
// MI455X (gfx1250) — compile-verified
//
#include <hip/hip_runtime.h>
#include <hip/hip_bf16.h>

#define NUM_LAYERS 4
#define OUT_DIM 16384
#define NUM_CLASSES 16
#define TAU 10.0f

typedef __attribute__((ext_vector_type(2))) float v2f;
typedef __attribute__((ext_vector_type(8))) float v8f;

// Per-lane precomputed B-matrix operand for V_WMMA_F32_16X16X4_F32.
// B tile for K-chunk i is coeff_table rows [4i+off, 4i+off+1] (off = lane>=16 ? 2:0),
// column N = lane&15 (zero for N>=4).  Btab[lane][2i] = B-vgpr0, [2i+1] = B-vgpr1.
__constant__ float c_btab[32][8] = {
    /*L0 */ {0, 0, 0, 0, 1, 1, 1, 1},
    /*L1 */ {0, 0, 0, 0, -1, -1, -1, -1},
    /*L2 */ {0, 0, 1, 1, -1, -1, 0, 0},
    /*L3 */ {0, 1, -1, 0, 1, 2, 0, 1},
    /*L4 */ {0, 0, 0, 0, 0, 0, 0, 0},
    /*L5 */ {0, 0, 0, 0, 0, 0, 0, 0},
    /*L6 */ {0, 0, 0, 0, 0, 0, 0, 0},
    /*L7 */ {0, 0, 0, 0, 0, 0, 0, 0},
    /*L8 */ {0, 0, 0, 0, 0, 0, 0, 0},
    /*L9 */ {0, 0, 0, 0, 0, 0, 0, 0},
    /*L10*/ {0, 0, 0, 0, 0, 0, 0, 0},
    /*L11*/ {0, 0, 0, 0, 0, 0, 0, 0},
    /*L12*/ {0, 0, 0, 0, 0, 0, 0, 0},
    /*L13*/ {0, 0, 0, 0, 0, 0, 0, 0},
    /*L14*/ {0, 0, 0, 0, 0, 0, 0, 0},
    /*L15*/ {0, 0, 0, 0, 0, 0, 0, 0},
    /*L16*/ {0, 0, 0, 0, 1, 1, 1, 1},
    /*L17*/ {1, 1, 1, 1, 0, 0, 0, 0},
    /*L18*/ {0, 0, 1, 1, -1, -1, 0, 0},
    /*L19*/ {-1, 0, -2, -1, 0, 1, -1, 0},
    /*L20*/ {0, 0, 0, 0, 0, 0, 0, 0},
    /*L21*/ {0, 0, 0, 0, 0, 0, 0, 0},
    /*L22*/ {0, 0, 0, 0, 0, 0, 0, 0},
    /*L23*/ {0, 0, 0, 0, 0, 0, 0, 0},
    /*L24*/ {0, 0, 0, 0, 0, 0, 0, 0},
    /*L25*/ {0, 0, 0, 0, 0, 0, 0, 0},
    /*L26*/ {0, 0, 0, 0, 0, 0, 0, 0},
    /*L27*/ {0, 0, 0, 0, 0, 0, 0, 0},
    /*L28*/ {0, 0, 0, 0, 0, 0, 0, 0},
    /*L29*/ {0, 0, 0, 0, 0, 0, 0, 0},
    /*L30*/ {0, 0, 0, 0, 0, 0, 0, 0},
    /*L31*/ {0, 0, 0, 0, 0, 0, 0, 0}};

// ---------------------------------------------------------------------------
// Kernel 1: c[l][j][0..3] = softmax(weights[l][j]) @ coeff_table
// One wave = 16 rows; K=16 via four chained V_WMMA_F32_16X16X4_F32.
// All register-array indices are compile-time constants -> no cndmask chains.
// ---------------------------------------------------------------------------
__global__ __launch_bounds__(256) void coeff_kernel(const float* __restrict__ weights,
                                                    float* __restrict__ c_out) {
  const int tid  = threadIdx.x;
  const int lane = tid & 31;
  const int wave = tid >> 5;
  const int row_base = blockIdx.x * 128 + wave * 16;  // 8 waves * 16 rows
  const int m = lane & 15;           // row-in-tile for A; column N for B/D
  const bool hi = (lane >= 16);
  const int r = row_base + m;

  // Row r's 16 weights via 4 x b128 (lanes L and L+16 read the same row).
  const float4* wr = (const float4*)(weights + (size_t)r * 16);
  const float4 w0 = wr[0], w1 = wr[1], w2 = wr[2], w3 = wr[3];
  float w[16] = {w0.x, w0.y, w0.z, w0.w, w1.x, w1.y, w1.z, w1.w,
                 w2.x, w2.y, w2.z, w2.w, w3.x, w3.y, w3.z, w3.w};

  float mx = w[0];
#pragma unroll
  for (int k = 1; k < 16; ++k) mx = fmaxf(mx, w[k]);
  float p[16];
  float s = 0.0f;
#pragma unroll
  for (int k = 0; k < 16; ++k) { p[k] = expf(w[k] - mx); s += p[k]; }
  const float inv = 1.0f / s;
#pragma unroll
  for (int k = 0; k < 16; ++k) p[k] *= inv;

  // A operand per chunk: lanes<16 carry K=4i,4i+1; lanes>=16 carry K=4i+2,4i+3.
  float pa[4], pb[4];
#pragma unroll
  for (int i = 0; i < 4; ++i) {
    pa[i] = hi ? p[4 * i + 2] : p[4 * i + 0];
    pb[i] = hi ? p[4 * i + 3] : p[4 * i + 1];
  }

  // B operand: two b128 loads from the precomputed per-lane table.
  const float4* bt = (const float4*)&c_btab[lane][0];
  const float4 bA = bt[0], bB = bt[1];
  const float bv[8] = {bA.x, bA.y, bA.z, bA.w, bB.x, bB.y, bB.z, bB.w};

  v8f acc = {};
#pragma unroll
  for (int i = 0; i < 4; ++i) {
    v2f a; a.x = pa[i];       a.y = pb[i];
    v2f b; b.x = bv[2 * i];   b.y = bv[2 * i + 1];
    // (neg_a, A, neg_b, B, c_mod, C, reuse_a, reuse_b)
    acc = __builtin_amdgcn_wmma_f32_16x16x4_f32(false, a, false, b, (short)0,
                                                acc, false, false);
  }

  // D layout: lane<16, vgpr v -> (M=v, N=lane); lane>=16 -> (M=v+8, N=lane-16).
  if (m < 4) {
    const int mb = row_base + (hi ? 8 : 0);
#pragma unroll
    for (int v = 0; v < 8; ++v) c_out[(size_t)(mb + v) * 4 + m] = acc[v];
  }
}

// ---------------------------------------------------------------------------
// Kernel 2: one workgroup per batch row.  Row loaded straight into LDS with
// async global->LDS DMA (ASYNCcnt path), then 4 gather layers ping-pong
// between two 64KB LDS buffers, then per-class reduction.  132KB LDS.
// ---------------------------------------------------------------------------
__global__ __launch_bounds__(256) void fuse_kernel(const float* __restrict__ x,
                                                   const int* __restrict__ a_idx,
                                                   const int* __restrict__ b_idx,
                                                   const float* __restrict__ c_all,
                                                   float* __restrict__ out) {
  __shared__ float buf0[OUT_DIM];
  __shared__ float buf1[OUT_DIM];
  __shared__ float red[256];

  const int tid = threadIdx.x;
  const int row = blockIdx.x;

  // Async copy of this batch row into LDS buf0: per-lane 16B global->LDS.
  const float4* xr  = (const float4*)(x + (size_t)row * OUT_DIM);
  float4*       b0v = (float4*)buf0;
#pragma unroll
  for (int v = 0; v < 16; ++v) {
    const int g = v * 256 + tid;                       // float4 index, 0..4095
    const unsigned lds_dst = (unsigned)(uintptr_t)(b0v + g);  // LDS byte offset
    const float4*  gsrc    = xr + g;
    asm volatile("global_load_async_to_lds_b128 %0, %1, off"
                 :
                 : "v"(lds_dst), "v"(gsrc)
                 : "memory");
  }
  asm volatile("s_wait_asynccnt 0x0" ::: "memory");
  __syncthreads();

  float* cur = buf0;
  float* nxt = buf1;
#pragma unroll
  for (int l = 0; l < NUM_LAYERS; ++l) {
    const int* ai = a_idx + l * OUT_DIM;
    const int* bi = b_idx + l * OUT_DIM;
    const float4* cl = (const float4*)(c_all + (size_t)l * OUT_DIM * 4);
#pragma unroll 8
    for (int v = 0; v < 64; ++v) {
      const int j = v * 256 + tid;      // coalesced index/coeff reads
      const int ja = ai[j];
      const int jb = bi[j];
      const float4 c = cl[j];
      const float a = cur[ja];          // ds_load_b32 gathers (LDS-resident)
      const float b = cur[jb];
      nxt[j] = c.x + c.y * a + c.z * b + c.w * (a * b);
    }
    __syncthreads();
    float* t = cur; cur = nxt; nxt = t;
  }

  // Per-class reduction: 16 threads per class, stride-16 within the class's
  // 1024-element segment, then a 16-wide finish.
  const int k = tid >> 4;
  const int off = tid & 15;
  float s = 0.0f;
  const float* base = cur + k * 1024 + off;
#pragma unroll
  for (int v = 0; v < 64; ++v) s += base[16 * v];
  red[tid] = s;
  __syncthreads();
  if (tid < NUM_CLASSES) {
    float t = 0.0f;
#pragma unroll
    for (int j = 0; j < 16; ++j) t += red[tid * 16 + j];
    out[row * NUM_CLASSES + tid] = t * (1.0f / TAU);
  }
}

extern "C" void kernel_launch(void* const* d_in, const int* in_sizes, int n_in,
                              void* d_out, int out_size, void* d_ws, size_t ws_size,
                              hipStream_t stream) {
  const float* x       = (const float*)d_in[0];   // (B, 16384) f32
  const float* weights = (const float*)d_in[1];   // (4, 16384, 16) f32
  const int*   a_idx   = (const int*)d_in[2];     // (4, 16384) i32
  const int*   b_idx   = (const int*)d_in[3];     // (4, 16384) i32
  float*       out     = (float*)d_out;           // (B, 16) f32
  float*       c_all   = (float*)d_ws;            // (4, 16384, 4) f32 = 1 MB scratch

  const int batch = in_sizes[0] / OUT_DIM;        // 512 in the reference

  coeff_kernel<<<(NUM_LAYERS * OUT_DIM) / 128, 256, 0, stream>>>(weights, c_all);
  fuse_kernel<<<batch, 256, 0, stream>>>(x, a_idx, b_idx, c_all, out);
}